// DepthToNormals_76794015252860
// MI455X (gfx1250) — compile-verified
//
#include <hip/hip_runtime.h>
#include <stdint.h>

// Problem constants (match reference)
#define BB 4
#define HH 512
#define WW 1024
#define TILE_W 32
#define TILE_H 8
#define HALO_W (TILE_W + 2)   // 34
#define HALO_H (TILE_H + 2)   // 10
#define NTHREADS 256

#if __has_builtin(__builtin_amdgcn_tensor_load_to_lds)
#define HAS_TDM 1
#endif

typedef unsigned int v4u __attribute__((ext_vector_type(4)));
typedef int          v4i __attribute__((ext_vector_type(4)));
typedef int          v8i __attribute__((ext_vector_type(8)));

__global__ __launch_bounds__(NTHREADS)
void DepthToNormals_kernel(const float* __restrict__ depth, float* __restrict__ out) {
    __shared__ float sh[HALO_H * HALO_W];          // depth halo tile (1360 B)
    __shared__ float s_sp[HALO_H], s_cp[HALO_H];   // sin/cos(phi) per tile row
    __shared__ float s_st[HALO_W], s_ct[HALO_W];   // sin/cos(theta) per tile col

    const int tid = threadIdx.x;
    const int tx  = blockIdx.x;
    const int ty  = blockIdx.y;
    const int bz  = blockIdx.z;
    const int gx0 = tx * TILE_W - 1;               // halo origin (may be -1)
    const int gy0 = ty * TILE_H - 1;
    const float* __restrict__ dimg = depth + (size_t)bz * HH * WW;

    // 91% of tiles are interior (1860/2048): make that the fall-through path.
    const bool interior = __builtin_expect(
        (tx > 0) && (ty > 0) && (tx < (WW / TILE_W - 1)) && (ty < (HH / TILE_H - 1)), 1);
    (void)interior;

    // ---- stage 1: fire the halo-tile copy into LDS (async where possible) ----
#ifdef HAS_TDM
    if (interior) {
        // Tensor Data Mover: DMA the 34x10 fp32 halo tile into LDS.
        // Issued by wave 0 FIRST, waited on LAST, so the DMA overlaps the
        // sincos ray-table computation below.
        if (tid < 32) {
            uint32_t lds_addr = (uint32_t)(size_t)(void*)sh;
            uint64_t ga = (uint64_t)(size_t)(const void*)(dimg + (size_t)gy0 * WW + gx0);

            // D# group 0: count=1 | lds_addr | global_addr[56:0] | type=2
            v4u g0;
            g0.x = 1u;
            g0.y = lds_addr;
            g0.z = (uint32_t)ga;
            g0.w = ((uint32_t)(ga >> 32) & 0x1FFFFFFu) | (2u << 30);

            // D# group 1: data_size=4B; tensor 1024x512, row stride 1024; tile 34x10
            v8i g1;
            g1[0] = (int)(2u << 16);                        // data_size=2 (4 bytes)
            g1[1] = (int)(((uint32_t)WW & 0xFFFFu) << 16);  // tensor_dim0[15:0]
            g1[2] = (int)(((uint32_t)HH & 0xFFFFu) << 16);  // dim0 hi=0 | tensor_dim1[15:0]
            g1[3] = (int)((uint32_t)HALO_W << 16);          // dim1 hi=0 | tile_dim0=34
            g1[4] = (int)(uint32_t)HALO_H;                  // tile_dim1=10 | tile_dim2=0
            g1[5] = (int)(uint32_t)WW;                      // tensor_dim0_stride[31:0]
            g1[6] = 0;
            g1[7] = 0;

            v4i z4 = {0, 0, 0, 0};                          // 2D tensor: groups 2/3 unused
#if __clang_major__ >= 23
            v8i z8 = {0, 0, 0, 0, 0, 0, 0, 0};
            __builtin_amdgcn_tensor_load_to_lds(g0, g1, z4, z4, z8, 0);
#else
            __builtin_amdgcn_tensor_load_to_lds(g0, g1, z4, z4, 0);
#endif
        }
    } else
#endif
    {
        // Border tiles (and non-TDM fallback): bounds-checked cooperative fill.
        // Zero fill == reference's zero-padded points (point = 0*ray = 0).
        for (int idx = tid; idx < HALO_H * HALO_W; idx += NTHREADS) {
            int r  = idx / HALO_W;
            int c  = idx - r * HALO_W;
            int gy = gy0 + r;
            int gx = gx0 + c;
            float v = 0.0f;
            if (gy >= 0 && gy < HH && gx >= 0 && gx < WW) v = dimg[(size_t)gy * WW + gx];
            sh[idx] = v;
        }
    }

    // ---- stage 2 (overlaps the TDM DMA): ray tables, 44 hardware sincos ----
    // Arguments are bounded (|phi| <= pi/2+eps, |theta| <= pi+eps), so the
    // hardware v_sin_f32/v_cos_f32 path (__sinf/__cosf) is accurate here and
    // avoids libm's Payne-Hanek huge-argument reduction.
    const float PI = 3.14159265358979323846f;
    if (tid < HALO_H) {
        float phi = 0.5f * PI - ((float)(gy0 + tid) + 0.5f) * (PI / (float)HH);
        s_sp[tid] = __sinf(phi);
        s_cp[tid] = __cosf(phi);
    }
    if (tid >= 64 && tid < 64 + HALO_W) {
        int c = tid - 64;
        float th = ((float)(gx0 + c) + 0.5f) * (2.0f * PI / (float)WW) - PI;
        s_st[c] = __sinf(th);
        s_ct[c] = __cosf(th);
    }

#ifdef HAS_TDM
    // Only the issuing wave carries TENSORcnt; everyone else syncs at the barrier.
    if (interior && tid < 32) {
        __builtin_amdgcn_s_wait_tensorcnt(0);
    }
#endif
    __syncthreads();

    // ---- per-pixel 3x3 Gram + analytic 3x3 solve (pure fp32 VALU) ----
    const int lx = tid & (TILE_W - 1);
    const int ly = tid >> 5;

    float sp[3] = { s_sp[ly],     s_sp[ly + 1], s_sp[ly + 2] };
    float cp[3] = { s_cp[ly],     s_cp[ly + 1], s_cp[ly + 2] };
    float st[3] = { s_st[lx],     s_st[lx + 1], s_st[lx + 2] };
    float ct[3] = { s_ct[lx],     s_ct[lx + 1], s_ct[lx + 2] };

    float G00 = 0.f, G01 = 0.f, G02 = 0.f, G11 = 0.f, G12 = 0.f, G22 = 0.f;
    float b0 = 0.f, b1 = 0.f, b2 = 0.f;
#pragma unroll
    for (int di = 0; di < 3; ++di) {
        float sphi = sp[di], cphi = cp[di];
#pragma unroll
        for (int dj = 0; dj < 3; ++dj) {
            float d  = sh[(ly + di) * HALO_W + (lx + dj)];
            float px = d * cphi * st[dj];
            float py = d * sphi;
            float pz = d * cphi * ct[dj];
            G00 += px * px; G01 += px * py; G02 += px * pz;
            G11 += py * py; G12 += py * pz; G22 += pz * pz;
            b0 += px; b1 += py; b2 += pz;
        }
    }
    G00 += 1e-5f; G11 += 1e-5f; G22 += 1e-5f;

    // adjugate (G symmetric)
    float a00 = G11 * G22 - G12 * G12;
    float a01 = G02 * G12 - G01 * G22;
    float a02 = G01 * G12 - G02 * G11;
    float a11 = G00 * G22 - G02 * G02;
    float a12 = G01 * G02 - G00 * G12;
    float a22 = G00 * G11 - G01 * G01;
    float det = G00 * a00 + G01 * a01 + G02 * a02;
    float idet = 1.0f / det;

    float n0 = (a00 * b0 + a01 * b1 + a02 * b2) * idet;
    float n1 = (a01 * b0 + a11 * b1 + a12 * b2) * idet;
    float n2 = (a02 * b0 + a12 * b1 + a22 * b2) * idet;

    float nrm = sqrtf(n0 * n0 + n1 * n1 + n2 * n2);
    nrm = (nrm == 0.0f) ? 1e-4f : nrm;     // norm + (1 - sign(|norm|))*1e-4
    float s = -1.0f / nrm;

    // Output is write-once streamed data: non-temporal stores keep it from
    // displacing depth tiles / ray tables in the cache hierarchy.
    const size_t plane = (size_t)HH * WW;
    const size_t o = (size_t)bz * 3 * plane
                   + (size_t)(ty * TILE_H + ly) * WW + (size_t)(tx * TILE_W + lx);
    __builtin_nontemporal_store(n0 * s, &out[o]);
    __builtin_nontemporal_store(n1 * s, &out[o + plane]);
    __builtin_nontemporal_store(n2 * s, &out[o + 2 * plane]);
}

extern "C" void kernel_launch(void* const* d_in, const int* in_sizes, int n_in,
                              void* d_out, int out_size, void* d_ws, size_t ws_size,
                              hipStream_t stream) {
    (void)in_sizes; (void)n_in; (void)out_size; (void)d_ws; (void)ws_size;
    const float* depth = (const float*)d_in[0];
    float* out = (float*)d_out;
    dim3 grid(WW / TILE_W, HH / TILE_H, BB);   // 32 x 64 x 4 = 8192 blocks
    DepthToNormals_kernel<<<grid, NTHREADS, 0, stream>>>(depth, out);
}